// Graph_Generator_71863392796894
// MI455X (gfx1250) — compile-verified
//
#include <hip/hip_runtime.h>
#include <math.h>

// MI455X / gfx1250, wave32. fp32 GEMMs via V_WMMA_F32_16X16X4_F32.

typedef float v2f __attribute__((ext_vector_type(2)));
typedef float v8f __attribute__((ext_vector_type(8)));

#define B_   8
#define T_   12
#define N_   2048
#define C_   60
#define S_   60
#define KSEL 1638          // int(2048 * 0.8)
#define NT   (N_ / 16)     // 128 row-tiles per batch

static __device__ __forceinline__ v8f wmma_f32(v2f a, v2f b, v8f c) {
    // 8 args: (neg_a, A, neg_b, B, c_mod, C, reuse_a, reuse_b)
    return __builtin_amdgcn_wmma_f32_16x16x4_f32(false, a, false, b, (short)0, c,
                                                 false, false);
}

// ---------------------------------------------------------------------------
// Kernel 1: xs[b,n,c] = sum_t x[b,t,n,c]
// ---------------------------------------------------------------------------
__global__ __launch_bounds__(256) void k_xsum(const float* __restrict__ x,
                                              float* __restrict__ xs) {
    int idx = blockIdx.x * 256 + threadIdx.x;
    if (idx >= B_ * N_ * C_) return;
    int c = idx % C_;
    int n = (idx / C_) % N_;
    int b = idx / (C_ * N_);
    float s = 0.f;
    #pragma unroll
    for (int t = 0; t < T_; ++t)
        s += x[(((size_t)(b * T_ + t)) * N_ + n) * C_ + c];
    xs[idx] = s;
}

// ---------------------------------------------------------------------------
// Kernel 2: ea_sum[b,n,s] = sum_t softmax_s( x[b,t,n,:] @ U1 )
// One wave32 per 16-row tile. WMMA M=16, N-tiles of 16 (pad 60->64), K=60.
// ---------------------------------------------------------------------------
__global__ __launch_bounds__(32) void k_attn_sum(const float* __restrict__ x,
                                                 const float* __restrict__ U1,
                                                 float* __restrict__ ea_sum) {
    __shared__ float xt[16 * C_];     // 16x60 input tile
    __shared__ float lg[16 * 64];     // 16x64 logits (cols >= 60 unused)
    __shared__ float eacc[16 * S_];   // softmax accumulator over t

    int bid  = blockIdx.x;
    int b    = bid >> 7;
    int n0   = (bid & 127) << 4;
    int lane = threadIdx.x;
    int half = lane >> 4;
    int mrow = lane & 15;

    for (int i = lane; i < 16 * S_; i += 32) eacc[i] = 0.f;
    __syncthreads();

    for (int t = 0; t < T_; ++t) {
        for (int i = lane; i < 16 * C_; i += 32) {
            int r = i / C_, c = i % C_;
            xt[i] = x[(((size_t)(b * T_ + t)) * N_ + n0 + r) * C_ + c];
        }
        __syncthreads();

        for (int st = 0; st < 64; st += 16) {
            int  col = st + mrow;
            bool cok = (col < S_);
            v8f acc = {0.f, 0.f, 0.f, 0.f, 0.f, 0.f, 0.f, 0.f};
            #pragma unroll
            for (int k0 = 0; k0 < C_; k0 += 4) {
                int kk = k0 + 2 * half;
                v2f a, bb;
                a.x  = xt[mrow * C_ + kk];
                a.y  = xt[mrow * C_ + kk + 1];
                bb.x = cok ? U1[kk * S_ + col] : 0.f;
                bb.y = cok ? U1[(kk + 1) * S_ + col] : 0.f;
                acc = wmma_f32(a, bb, acc);
            }
            #pragma unroll
            for (int r = 0; r < 8; ++r)
                lg[(r + 8 * half) * 64 + col] = acc[r];
        }
        __syncthreads();

        if (lane < 16) {            // one lane per row: stable softmax + accumulate
            int row = lane;
            float mx = -1e30f;
            for (int s = 0; s < S_; ++s) mx = fmaxf(mx, lg[row * 64 + s]);
            float sum = 0.f;
            for (int s = 0; s < S_; ++s) sum += __expf(lg[row * 64 + s] - mx);
            float inv = 1.f / sum;
            for (int s = 0; s < S_; ++s)
                eacc[row * S_ + s] += __expf(lg[row * 64 + s] - mx) * inv;
        }
        __syncthreads();
    }

    for (int i = lane; i < 16 * S_; i += 32) {
        int r = i / S_, s = i % S_;
        ea_sum[((size_t)(b * N_) + n0 + r) * S_ + s] = eacc[i];
    }
}

// ---------------------------------------------------------------------------
// Kernel 3: ex_sum = ea_sum @ U2   (16384x60 @ 60x60, fp32 WMMA)
// ---------------------------------------------------------------------------
__global__ __launch_bounds__(32) void k_ex(const float* __restrict__ ea,
                                           const float* __restrict__ U2,
                                           float* __restrict__ ex) {
    __shared__ float at[16 * S_];
    int bid  = blockIdx.x;
    int b    = bid >> 7;
    int n0   = (bid & 127) << 4;
    int lane = threadIdx.x;
    int half = lane >> 4;
    int mrow = lane & 15;

    for (int i = lane; i < 16 * S_; i += 32) {
        int r = i / S_, s = i % S_;
        at[i] = ea[((size_t)(b * N_) + n0 + r) * S_ + s];
    }
    __syncthreads();

    for (int st = 0; st < 64; st += 16) {
        int  col = st + mrow;
        bool cok = (col < C_);
        v8f acc = {0.f, 0.f, 0.f, 0.f, 0.f, 0.f, 0.f, 0.f};
        #pragma unroll
        for (int k0 = 0; k0 < S_; k0 += 4) {
            int kk = k0 + 2 * half;
            v2f a, bb;
            a.x  = at[mrow * S_ + kk];
            a.y  = at[mrow * S_ + kk + 1];
            bb.x = cok ? U2[kk * C_ + col] : 0.f;
            bb.y = cok ? U2[(kk + 1) * C_ + col] : 0.f;
            acc = wmma_f32(a, bb, acc);
        }
        if (cok) {
            #pragma unroll
            for (int r = 0; r < 8; ++r)
                ex[((size_t)(b * N_) + n0 + r + 8 * half) * C_ + col] = acc[r];
        }
    }
}

// ---------------------------------------------------------------------------
// Kernel 4: fused adjacency scores.
//   r1 = xs[b,n,:].spatial[m,:],  r2 = ex[b,n,:].ea[b,m,:]
//   e  = exp(relu(r/sqrt(T))),  V = w0*e1/Z1 + w1*e2/Z2 (+bias)
// Two passes (Z sums, then values), scores written into d_out as scratch.
// ---------------------------------------------------------------------------
static __device__ __forceinline__ void adj_tile(const float* xt, const float* et,
                                                const float* __restrict__ sp,
                                                const float* __restrict__ eaB,
                                                int mrow, int half, int mt,
                                                v8f& a1, v8f& a2) {
    v8f z = {0.f, 0.f, 0.f, 0.f, 0.f, 0.f, 0.f, 0.f};
    a1 = z; a2 = z;
    int m = mt + mrow;
    #pragma unroll
    for (int k0 = 0; k0 < C_; k0 += 4) {
        int kk = k0 + 2 * half;
        v2f a, bb;
        a.x  = xt[mrow * C_ + kk];
        a.y  = xt[mrow * C_ + kk + 1];
        bb.x = sp[(size_t)m * C_ + kk];
        bb.y = sp[(size_t)m * C_ + kk + 1];
        a1 = wmma_f32(a, bb, a1);
        a.x  = et[mrow * C_ + kk];
        a.y  = et[mrow * C_ + kk + 1];
        bb.x = eaB[(size_t)m * S_ + kk];
        bb.y = eaB[(size_t)m * S_ + kk + 1];
        a2 = wmma_f32(a, bb, a2);
    }
}

__global__ __launch_bounds__(32) void k_adj(const float* __restrict__ xs,
                                            const float* __restrict__ ea,
                                            const float* __restrict__ ex,
                                            const float* __restrict__ sp,
                                            const float* __restrict__ fcw,
                                            const float* __restrict__ fcb,
                                            float* __restrict__ V) {
    __shared__ float xt[16 * C_];
    __shared__ float et[16 * C_];
    __shared__ float z1[16], z2[16];

    int bid  = blockIdx.x;
    int b    = bid >> 7;
    int n0   = (bid & 127) << 4;
    int lane = threadIdx.x;
    int half = lane >> 4;
    int mrow = lane & 15;
    const float invS = 0.28867513459481287f;   // 1/sqrt(12)
    const float* eaB = ea + (size_t)b * N_ * S_;

    for (int i = lane; i < 16 * C_; i += 32) {
        int r = i / C_, c = i % C_;
        xt[i] = xs[((size_t)(b * N_) + n0 + r) * C_ + c];
        et[i] = ex[((size_t)(b * N_) + n0 + r) * C_ + c];
    }
    __syncthreads();

    // ---- pass 1: Z1, Z2 row sums ----
    float p1[8], p2[8];
    #pragma unroll
    for (int r = 0; r < 8; ++r) { p1[r] = 0.f; p2[r] = 0.f; }
    for (int mt = 0; mt < N_; mt += 16) {
        v8f a1, a2;
        adj_tile(xt, et, sp, eaB, mrow, half, mt, a1, a2);
        #pragma unroll
        for (int r = 0; r < 8; ++r) {
            p1[r] += __expf(fmaxf(a1[r] * invS, 0.f));
            p2[r] += __expf(fmaxf(a2[r] * invS, 0.f));
        }
    }
    #pragma unroll
    for (int off = 1; off < 16; off <<= 1) {   // butterfly within each 16-lane half
        #pragma unroll
        for (int r = 0; r < 8; ++r) {
            p1[r] += __shfl_xor(p1[r], off, 32);
            p2[r] += __shfl_xor(p2[r], off, 32);
        }
    }
    if (mrow == 0) {
        #pragma unroll
        for (int r = 0; r < 8; ++r) { z1[8 * half + r] = p1[r]; z2[8 * half + r] = p2[r]; }
    }
    __syncthreads();

    float w0 = fcw[0], w1 = fcw[1], bias = fcb[0];
    float iz1[8], iz2[8];
    #pragma unroll
    for (int r = 0; r < 8; ++r) {
        iz1[r] = w0 / z1[8 * half + r];
        iz2[r] = w1 / z2[8 * half + r];
    }

    // ---- pass 2: combined scores -> d_out (scratch) ----
    for (int mt = 0; mt < N_; mt += 16) {
        v8f a1, a2;
        adj_tile(xt, et, sp, eaB, mrow, half, mt, a1, a2);
        #pragma unroll
        for (int r = 0; r < 8; ++r) {
            int row = 8 * half + r;
            float v = iz1[r] * __expf(fmaxf(a1[r] * invS, 0.f)) +
                      iz2[r] * __expf(fmaxf(a2[r] * invS, 0.f)) + bias;
            V[((size_t)(b * N_) + n0 + row) * N_ + mt + mrow] = v;
        }
    }
}

// ---------------------------------------------------------------------------
// Kernel 5: per-row top-k (k=1638) binary mask, in-place over d_out.
// MSB radix-select on order-preserving uint32 keys; ties broken lowest-index
// first (matches jax.lax.top_k) via chunked prefix scan.
// ---------------------------------------------------------------------------
__global__ __launch_bounds__(256) void k_topk(float* __restrict__ V) {
    __shared__ unsigned keys[N_];
    __shared__ int cnt;
    __shared__ int chunk[256];

    int tid = threadIdx.x;
    size_t base = (size_t)blockIdx.x * N_;

    for (int i = tid; i < N_; i += 256) {
        unsigned b = __float_as_uint(V[base + i]);
        keys[i] = (b & 0x80000000u) ? ~b : (b | 0x80000000u);
    }
    __syncthreads();

    unsigned prefix = 0;
    for (int bit = 31; bit >= 0; --bit) {
        unsigned cand = prefix | (1u << bit);
        if (tid == 0) cnt = 0;
        __syncthreads();
        int lc = 0;
        for (int i = tid; i < N_; i += 256) lc += (keys[i] >= cand) ? 1 : 0;
        atomicAdd(&cnt, lc);
        __syncthreads();
        if (cnt >= KSEL) prefix = cand;   // uniform across block
        __syncthreads();
    }
    unsigned Tk = prefix;                 // k-th largest key

    if (tid == 0) cnt = 0;
    __syncthreads();
    int lg = 0;
    for (int i = tid; i < N_; i += 256) lg += (keys[i] > Tk) ? 1 : 0;
    atomicAdd(&cnt, lg);
    __syncthreads();
    int ties = KSEL - cnt;                // how many ==Tk elements to keep

    int c0 = tid * 8;                     // contiguous chunk of 8 per thread
    int myc = 0;
    #pragma unroll
    for (int j = 0; j < 8; ++j) myc += (keys[c0 + j] == Tk) ? 1 : 0;
    chunk[tid] = myc;
    __syncthreads();
    for (int off = 1; off < 256; off <<= 1) {   // inclusive Hillis-Steele scan
        int v = (tid >= off) ? chunk[tid - off] : 0;
        __syncthreads();
        chunk[tid] += v;
        __syncthreads();
    }
    int rank = chunk[tid] - myc;          // exclusive prefix of equals

    #pragma unroll
    for (int j = 0; j < 8; ++j) {
        unsigned kk = keys[c0 + j];
        float o;
        if (kk > Tk)       o = 1.f;
        else if (kk == Tk) { o = (rank < ties) ? 1.f : 0.f; ++rank; }
        else               o = 0.f;
        V[base + c0 + j] = o;
    }
}

// ---------------------------------------------------------------------------
extern "C" void kernel_launch(void* const* d_in, const int* in_sizes, int n_in,
                              void* d_out, int out_size, void* d_ws, size_t ws_size,
                              hipStream_t stream) {
    (void)in_sizes; (void)n_in; (void)out_size; (void)ws_size;
    const float* x   = (const float*)d_in[0];
    const float* U1  = (const float*)d_in[1];
    const float* U2  = (const float*)d_in[2];
    const float* sp  = (const float*)d_in[3];
    const float* fcw = (const float*)d_in[4];
    const float* fcb = (const float*)d_in[5];
    float* out = (float*)d_out;

    float* xs = (float*)d_ws;                       // [B,N,C]
    float* ea = xs + (size_t)B_ * N_ * C_;          // [B,N,S]
    float* ex = ea + (size_t)B_ * N_ * S_;          // [B,N,C]  (~12 MB total)

    k_xsum    <<<(B_ * N_ * C_ + 255) / 256, 256, 0, stream>>>(x, xs);
    k_attn_sum<<<B_ * NT, 32, 0, stream>>>(x, U1, ea);
    k_ex      <<<B_ * NT, 32, 0, stream>>>(ea, U2, ex);
    k_adj     <<<B_ * NT, 32, 0, stream>>>(xs, ea, ex, sp, fcw, fcb, out);
    k_topk    <<<B_ * N_, 256, 0, stream>>>(out);
}